// LocalAttention_38508676776079
// MI455X (gfx1250) — compile-verified
//
#include <hip/hip_runtime.h>

// LocalAttention, B=4 H=16 L=2048 D=64, R=256.
// Mask keeps only j >= i + 1793  =>
//   rows 0..254   : softmax over keys [i+1793, 2047]   (tiny WMMA region)
//   rows 255..2047: uniform attention = mean(v) over all keys
//
// Memory-bound problem (~67MB min traffic @ 23.3 TB/s). Four kernels:
//   1) split-L vmean partial reduction (float4 loads, 512 blocks for
//      chip-wide load concurrency)
//   2) combine partials -> vmean
//   3) float4 broadcast of vmean into every output row
//   4) fp32-WMMA flash attention over the 256x256 unmasked corner,
//      overwriting rows 0..254.

typedef __attribute__((ext_vector_type(2))) float v2f;
typedef __attribute__((ext_vector_type(8))) float v8f;

#define B_    4
#define H_    16
#define L_    2048
#define D_    64
#define RWIN  256
#define KEY0  (L_ - RWIN)       // 1792: first key tile row
#define DIAG  (L_ - RWIN + 1)   // 1793: unmasked iff j - i >= DIAG
#define LASTQ (L_ - 1 - DIAG)   // 254 : last query row with any unmasked key
#define NSPLIT 8                // L-splits for the vmean reduction
#define ROWS_PER_SPLIT (L_ / NSPLIT)   // 256

// ---------------------------------------------------------------------------
// Kernel 1: partial[bh*NSPLIT + s][d] = sum over 256 rows of v[bh][row][d]
// 512 blocks x 256 threads; float4 (b128) coalesced loads.
// Thread t: c = t%16 (float4 column), g = t/16 (row group of 16).
// ---------------------------------------------------------------------------
__global__ __launch_bounds__(256)
void vpartial_kernel(const float* __restrict__ v, float* __restrict__ partial) {
  __shared__ float4 red[256];
  const int blk = blockIdx.x;            // bh*NSPLIT + s
  const int bh  = blk >> 3;
  const int s   = blk & (NSPLIT - 1);
  const int t   = threadIdx.x;
  const int c   = t & 15;                // float4 column 0..15 (d = 4c..4c+3)
  const int g   = t >> 4;                // row group 0..15

  const float4* vb4 = (const float4*)(v + (size_t)bh * L_ * D_);
  const int row0 = s * ROWS_PER_SPLIT + g;

  float4 acc = make_float4(0.f, 0.f, 0.f, 0.f);
#pragma unroll
  for (int it = 0; it < ROWS_PER_SPLIT / 16; ++it) {
    float4 x = vb4[(size_t)(row0 + it * 16) * (D_ / 4) + c];
    acc.x += x.x; acc.y += x.y; acc.z += x.z; acc.w += x.w;
  }
  red[t] = acc;
  __syncthreads();
  // fold the 16 row groups (stride in t-space is 16 per group)
  if (t < 128) { float4 o = red[t + 128];
    red[t].x += o.x; red[t].y += o.y; red[t].z += o.z; red[t].w += o.w; }
  __syncthreads();
  if (t < 64)  { float4 o = red[t + 64];
    red[t].x += o.x; red[t].y += o.y; red[t].z += o.z; red[t].w += o.w; }
  __syncthreads();
  if (t < 32)  { float4 o = red[t + 32];
    red[t].x += o.x; red[t].y += o.y; red[t].z += o.z; red[t].w += o.w; }
  __syncthreads();
  if (t < 16) { // g==0 now holds sum of groups; one float4 per column c=t
    float4 o = red[t + 16];
    float4 r = red[t];
    r.x += o.x; r.y += o.y; r.z += o.z; r.w += o.w;
    ((float4*)partial)[(size_t)blk * (D_ / 4) + t] = r;
  }
}

// ---------------------------------------------------------------------------
// Kernel 2: vmean[bh][d] = (1/L) * sum_s partial[bh*NSPLIT+s][d]
// ---------------------------------------------------------------------------
__global__ __launch_bounds__(64)
void vcombine_kernel(const float* __restrict__ partial, float* __restrict__ vmean) {
  const int bh = blockIdx.x;
  const int d  = threadIdx.x;
  float s = 0.f;
#pragma unroll
  for (int sp = 0; sp < NSPLIT; ++sp)
    s += partial[(size_t)(bh * NSPLIT + sp) * D_ + d];
  vmean[bh * D_ + d] = s * (1.0f / (float)L_);
}

// ---------------------------------------------------------------------------
// Kernel 3: out[bh][i][:] = vmean[bh][:] for all rows (rows 0..254 are
// overwritten by the attention kernel afterwards). float4 stores.
// ---------------------------------------------------------------------------
__global__ __launch_bounds__(256)
void bcast_kernel(const float* __restrict__ vmean, float* __restrict__ out) {
  const size_t idx = (size_t)blockIdx.x * blockDim.x + threadIdx.x; // float4 id
  const size_t e   = idx * 4;
  const int bh = (int)(e / (size_t)(L_ * D_));
  const int d  = (int)(e & (size_t)(D_ - 1));
  const float4* vm4 = (const float4*)vmean;
  float4 val = vm4[bh * (D_ / 4) + (d >> 2)];
  ((float4*)out)[idx] = val;
}

// ---------------------------------------------------------------------------
// Kernel 4: fp32 WMMA flash attention over the unmasked corner.
// grid = B*H blocks, 512 threads = 16 waves; wave w owns query rows
// [16w, 16w+15]; streams 16 key tiles covering keys [1792, 2047].
//
// WMMA f32 16x16x4 operand layouts (ISA 7.12.2):
//   A (16x4): lane%16 = M row; V0/V1 = K {k0,k0+1} (lanes 0-15) or
//             {k0+2,k0+3} (lanes 16-31)
//   B (4x16): lane%16 = N col; same K split across V0/V1
//   C/D     : lane%16 = N col; v8f elem r -> M = r + 8*(lane/16)
// ---------------------------------------------------------------------------
__global__ __launch_bounds__(512)
void attn_kernel(const float* __restrict__ q, const float* __restrict__ k,
                 const float* __restrict__ v, float* __restrict__ out) {
  __shared__ float pLds[16][16 * 17];   // per-wave 16x16 P tile, stride 17

  const int bh   = blockIdx.x;          // 0..63
  const int wave = threadIdx.x >> 5;    // 0..15 -> query tile
  const int lane = threadIdx.x & 31;
  const int tcol = lane & 15;           // A row / B,C,D column index
  const int half = lane >> 4;           // K-pair / row-half selector

  const float* qb = q + (size_t)bh * L_ * D_ + (size_t)wave * 16 * D_;
  const float* kb = k + (size_t)bh * L_ * D_ + (size_t)KEY0 * D_;
  const float* vb = v + (size_t)bh * L_ * D_ + (size_t)KEY0 * D_;
  float*       ob = out + (size_t)bh * L_ * D_;

  // Preload this wave's Q tile as WMMA-A fragments, pre-scaled by 1/sqrt(D).
  v2f aq[16];
#pragma unroll
  for (int c4 = 0; c4 < 16; ++c4) {
    const float* p = qb + tcol * D_ + c4 * 4 + 2 * half;
    v2f a; a.x = p[0] * 0.125f; a.y = p[1] * 0.125f;
    aq[c4] = a;
  }

  const v8f vzero = {0.f, 0.f, 0.f, 0.f, 0.f, 0.f, 0.f, 0.f};
  v8f acc[4];                            // O accumulators: 4 tiles of d
  acc[0] = vzero; acc[1] = vzero; acc[2] = vzero; acc[3] = vzero;
  float mrow[8], lsum[8];
#pragma unroll
  for (int r = 0; r < 8; ++r) { mrow[r] = -3.0e38f; lsum[r] = 0.0f; }

  float* myP = &pLds[wave][0];

  for (int kt = 0; kt < 16; ++kt) {
    // ---- S = (Q/8) * K^T for this 16x16 tile ----
    v8f s = vzero;
#pragma unroll
    for (int c4 = 0; c4 < 16; ++c4) {
      const float* kp = kb + (kt * 16 + tcol) * D_ + c4 * 4 + 2 * half;
      v2f b; b.x = kp[0]; b.y = kp[1];
      s = __builtin_amdgcn_wmma_f32_16x16x4_f32(
              false, aq[c4], false, b, (short)0, s, false, false);
    }

    // ---- mask + online softmax (16-lane row reductions) ----
    const int j = KEY0 + kt * 16 + tcol;
    float alpha[8];
#pragma unroll
    for (int r = 0; r < 8; ++r) {
      const int i = wave * 16 + r + 8 * half;
      float sv = (j - i >= DIAG) ? s[r] : -10000.0f;
      float rm = sv;
      rm = fmaxf(rm, __shfl_xor(rm, 1));
      rm = fmaxf(rm, __shfl_xor(rm, 2));
      rm = fmaxf(rm, __shfl_xor(rm, 4));
      rm = fmaxf(rm, __shfl_xor(rm, 8));
      const float mnew = fmaxf(mrow[r], rm);
      alpha[r] = __expf(mrow[r] - mnew);
      const float p = __expf(sv - mnew);
      float ps = p;
      ps += __shfl_xor(ps, 1);
      ps += __shfl_xor(ps, 2);
      ps += __shfl_xor(ps, 4);
      ps += __shfl_xor(ps, 8);
      lsum[r] = lsum[r] * alpha[r] + ps;
      mrow[r] = mnew;
      myP[(r + 8 * half) * 17 + tcol] = p;   // stage P (C-layout -> LDS)
    }

    // ---- rescale O by alpha ----
#pragma unroll
    for (int nt = 0; nt < 4; ++nt)
#pragma unroll
      for (int r = 0; r < 8; ++r)
        acc[nt][r] *= alpha[r];

    // ---- O += P * Vtile  (A = P via LDS relayout, B = V rows) ----
#pragma unroll
    for (int nt = 0; nt < 4; ++nt) {
#pragma unroll
      for (int c4 = 0; c4 < 4; ++c4) {
        const int kk = c4 * 4 + 2 * half;
        v2f a; a.x = myP[tcol * 17 + kk];
               a.y = myP[tcol * 17 + kk + 1];
        const float* vp = vb + (kt * 16 + kk) * D_ + nt * 16 + tcol;
        v2f b; b.x = vp[0]; b.y = vp[D_];
        acc[nt] = __builtin_amdgcn_wmma_f32_16x16x4_f32(
                      false, a, false, b, (short)0, acc[nt], false, false);
      }
    }
  }

  // ---- epilogue: O / lsum, store rows that have unmasked keys ----
#pragma unroll
  for (int r = 0; r < 8; ++r) {
    const int i = wave * 16 + r + 8 * half;
    if (i <= LASTQ) {
      const float inv = 1.0f / lsum[r];
#pragma unroll
      for (int nt = 0; nt < 4; ++nt)
        ob[(size_t)i * D_ + nt * 16 + tcol] = acc[nt][r] * inv;
    }
  }
}

// ---------------------------------------------------------------------------
extern "C" void kernel_launch(void* const* d_in, const int* in_sizes, int n_in,
                              void* d_out, int out_size, void* d_ws, size_t ws_size,
                              hipStream_t stream) {
  const float* q = (const float*)d_in[0];
  const float* k = (const float*)d_in[1];
  const float* v = (const float*)d_in[2];
  float* out = (float*)d_out;

  float* partial = (float*)d_ws;                       // 64*8*64 floats = 128 KB
  float* vmean   = partial + B_ * H_ * NSPLIT * D_;    // 64*64 floats  = 16 KB

  vpartial_kernel<<<B_ * H_ * NSPLIT, 256, 0, stream>>>(v, partial);
  vcombine_kernel<<<B_ * H_, 64, 0, stream>>>(partial, vmean);

  const int n4 = (B_ * H_ * L_ * D_) / 4;              // float4 elements
  bcast_kernel<<<n4 / 256, 256, 0, stream>>>(vmean, out);

  attn_kernel<<<B_ * H_, 512, 0, stream>>>(q, k, v, out);
}